// GIN_4303557231211
// MI455X (gfx1250) — compile-verified
//
#include <hip/hip_runtime.h>
#include <stdint.h>

// ---------------------------------------------------------------------------
// GIN (3x GINConv + linear classifier) for MI455X (gfx1250, wave32).
//
// Roofline: scatter-add dominates (82M f32 atomics, all L2-resident);
// GEMMs are ~1 GFLOP total -> use precision-exact V_WMMA_F32_16X16X4_F32.
// ---------------------------------------------------------------------------

typedef float v2f __attribute__((ext_vector_type(2)));
typedef float v8f __attribute__((ext_vector_type(8)));

#define NN     10000
#define NF     128
#define NC     40
#define NEDGE  640000
#define MTILES (NN / 16)     // 625 (exact)
#define LDSPAD 132           // 128 + 4 floats: row stride shifts 4 banks/row

// ---------------------------------------------------------------------------
// t = h  (vectorized copy; destination for the edge accumulation)
// ---------------------------------------------------------------------------
__global__ __launch_bounds__(256) void gin_init_t(const float* __restrict__ h,
                                                  float* __restrict__ t, int n4) {
  int i = blockIdx.x * blockDim.x + threadIdx.x;
  if (i < n4) ((float4*)t)[i] = ((const float4*)h)[i];
}

// ---------------------------------------------------------------------------
// One wave32 per edge. src/dst are wave-uniform -> scalar loads.
// Lane l moves feats [4l,4l+3] via a single b128 load + 4 relaxed
// agent-scope f32 atomic adds (lowers to global_atomic_add_f32, no CAS loop).
// ---------------------------------------------------------------------------
__global__ __launch_bounds__(256) void gin_scatter(const float* __restrict__ h,
                                                   const long long* __restrict__ src,
                                                   const long long* __restrict__ dst,
                                                   float* __restrict__ t) {
  int gid  = blockIdx.x * blockDim.x + threadIdx.x;
  int edge = gid >> 5;
  int lane = gid & 31;
  if (edge >= NEDGE) return;
  int s = (int)src[edge];
  int d = (int)dst[edge];
  const float4 v = *(const float4*)(h + (size_t)s * NF + lane * 4);
  float* o = t + (size_t)d * NF + lane * 4;
  __hip_atomic_fetch_add(o + 0, v.x, __ATOMIC_RELAXED, __HIP_MEMORY_SCOPE_AGENT);
  __hip_atomic_fetch_add(o + 1, v.y, __ATOMIC_RELAXED, __HIP_MEMORY_SCOPE_AGENT);
  __hip_atomic_fetch_add(o + 2, v.z, __ATOMIC_RELAXED, __HIP_MEMORY_SCOPE_AGENT);
  __hip_atomic_fetch_add(o + 3, v.w, __ATOMIC_RELAXED, __HIP_MEMORY_SCOPE_AGENT);
}

// ---------------------------------------------------------------------------
// out[m,n] = relu( (T @ W^T)[m,n] + b[n] ) * clamp(mask[n],0,1)
// Block = 8 waves; block owns one 16-row M-tile (staged in LDS, 8x reuse),
// wave w owns N-tile w. K=128 -> 32 x v_wmma_f32_16x16x4_f32 per wave.
//
// F32 WMMA A/B fragment layout (ISA 7.12.2, 16x4):
//   lanes 0-15 : row/col = lane,   vgpr0=K+0, vgpr1=K+1
//   lanes 16-31: row/col = lane-16,vgpr0=K+2, vgpr1=K+3
// C/D: lane -> N, vgpr v -> M = v + 8*(lane>=16).
// ---------------------------------------------------------------------------
__global__ __launch_bounds__(256) void gin_gemm_relu(const float* __restrict__ T,
                                                     const float* __restrict__ W,
                                                     const float* __restrict__ bias,
                                                     const float* __restrict__ mask,
                                                     float* __restrict__ out) {
  __shared__ float As[16 * LDSPAD];
  const int m0 = blockIdx.x * 16;

  // cooperative A-tile stage: 16x128 floats, 8 per thread
  for (int i = threadIdx.x; i < 16 * NF; i += 256) {
    int r = i >> 7, c = i & (NF - 1);
    As[r * LDSPAD + c] = T[(size_t)(m0 + r) * NF + c];
  }
  __syncthreads();

  const int wave = threadIdx.x >> 5;      // 0..7 -> N-tile
  const int lane = threadIdx.x & 31;
  const int half = lane >> 4;             // 0 or 1
  const int r    = lane & 15;
  const int n0   = wave * 16;
  const int n    = n0 + r;

  v8f acc = {};
#pragma unroll
  for (int k = 0; k < NF; k += 4) {
    v2f a = *(const v2f*)&As[r * LDSPAD + k + 2 * half];                 // A[m0+r][k..]
    v2f b = *(const v2f*)&W[(size_t)n * NF + k + 2 * half];              // B[k][n] = W[n][k]
    acc = __builtin_amdgcn_wmma_f32_16x16x4_f32(false, a, false, b,
                                                (short)0, acc, false, false);
  }

  const float bv = bias[n];
  const float mk = fminf(fmaxf(mask[n], 0.0f), 1.0f);
#pragma unroll
  for (int v = 0; v < 8; ++v) {
    int m = m0 + v + 8 * half;
    float y = acc[v] + bv;
    out[(size_t)m * NF + n] = fmaxf(y, 0.0f) * mk;
  }
}

// ---------------------------------------------------------------------------
// Classifier: out[m,n] = (H @ Wl^T)[m,n] + bl[n], n < 40.
// 3 waves -> 3 N-tiles (48 cols). For n >= 40 we load a clamped Wl row so
// EXEC stays all-1s across the WMMA (ISA restriction); those columns are
// garbage and simply never stored.
// ---------------------------------------------------------------------------
__global__ __launch_bounds__(96) void gin_gemm_final(const float* __restrict__ H,
                                                     const float* __restrict__ Wl,
                                                     const float* __restrict__ bl,
                                                     float* __restrict__ out) {
  __shared__ float As[16 * LDSPAD];
  const int m0 = blockIdx.x * 16;

  for (int i = threadIdx.x; i < 16 * NF; i += 96) {
    int r = i >> 7, c = i & (NF - 1);
    As[r * LDSPAD + c] = H[(size_t)(m0 + r) * NF + c];
  }
  __syncthreads();

  const int wave = threadIdx.x >> 5;      // 0..2
  const int lane = threadIdx.x & 31;
  const int half = lane >> 4;
  const int r    = lane & 15;
  const int n    = wave * 16 + r;
  const int ncl  = (n < NC) ? n : (NC - 1);   // EXEC-safe clamped load row

  v8f acc = {};
#pragma unroll
  for (int k = 0; k < NF; k += 4) {
    v2f a = *(const v2f*)&As[r * LDSPAD + k + 2 * half];
    v2f b = *(const v2f*)&Wl[(size_t)ncl * NF + k + 2 * half];
    acc = __builtin_amdgcn_wmma_f32_16x16x4_f32(false, a, false, b,
                                                (short)0, acc, false, false);
  }

  if (n < NC) {
    const float bv = bl[n];
#pragma unroll
    for (int v = 0; v < 8; ++v) {
      int m = m0 + v + 8 * half;
      out[(size_t)m * NC + n] = acc[v] + bv;
    }
  }
}

// ---------------------------------------------------------------------------
// Orchestration. Inputs (reference order):
// 0:x 1:edge_index(int64,[2,E]) 2:W0 3:b0 4:W1 5:b1 6:W2 7:b2 8:p_drop 9:Wl 10:bl
// ---------------------------------------------------------------------------
extern "C" void kernel_launch(void* const* d_in, const int* in_sizes, int n_in,
                              void* d_out, int out_size, void* d_ws, size_t ws_size,
                              hipStream_t stream) {
  (void)in_sizes; (void)n_in; (void)out_size; (void)ws_size;

  const float*     x    = (const float*)d_in[0];
  const long long* ei   = (const long long*)d_in[1];
  const long long* src  = ei;
  const long long* dst  = ei + NEDGE;
  const float*     Wk[3] = {(const float*)d_in[2], (const float*)d_in[4], (const float*)d_in[6]};
  const float*     bk[3] = {(const float*)d_in[3], (const float*)d_in[5], (const float*)d_in[7]};
  const float*     mask = (const float*)d_in[8];
  const float*     Wl   = (const float*)d_in[9];
  const float*     bl   = (const float*)d_in[10];
  float*           out  = (float*)d_out;

  float* hbuf = (float*)d_ws;                       // [NN*NF]
  float* tbuf = hbuf + (size_t)NN * NF;             // [NN*NF]

  const int n4          = NN * NF / 4;              // 320000
  const int copyBlocks  = n4 / 256;                 // 1250
  const int scatBlocks  = (NEDGE * 32) / 256;       // 80000

  const float* h = x;
  for (int layer = 0; layer < 3; ++layer) {
    gin_init_t  <<<copyBlocks, 256, 0, stream>>>(h, tbuf, n4);
    gin_scatter <<<scatBlocks, 256, 0, stream>>>(h, src, dst, tbuf);
    gin_gemm_relu<<<MTILES, 256, 0, stream>>>(tbuf, Wk[layer], bk[layer], mask, hbuf);
    h = hbuf;
  }
  gin_gemm_final<<<MTILES, 96, 0, stream>>>(hbuf, Wl, bl, out);
}